// HalftoneMARLLoss_51531017617983
// MI455X (gfx1250) — compile-verified
//
#include <hip/hip_runtime.h>
#include <math.h>

// ---------------------------------------------------------------------------
// HalftoneMARLLoss for MI455X (gfx1250, wave32).
// Exact-math restructuring: single-pixel-flip candidates evaluated as rank-1
// updates of the blurred fields; Gaussian blur = G*X*G^T via f32 WMMA 16x16x4.
// ---------------------------------------------------------------------------

#define IMG    32
#define HWPIX  1024          // IMG*IMG
#define BATCH  2
#define KRAD   5
#define KSIZE  11
#define EPSF   1e-12f
#define C1F    1e-4f         // (0.01)^2
#define C2F    9e-4f         // (0.03)^2
#define WSF    0.06f

typedef float v2f __attribute__((ext_vector_type(2)));
typedef float v8f __attribute__((ext_vector_type(8)));

struct Gauss { float g[KSIZE]; };

// ------------------------- threefry2x32 (JAX PRNG) -------------------------
__device__ __forceinline__ unsigned rotl32(unsigned x, int r) {
  return (x << r) | (x >> (32 - r));
}

__device__ void threefry2x32(unsigned k0, unsigned k1, unsigned x0, unsigned x1,
                             unsigned* o0, unsigned* o1) {
  unsigned ks0 = k0, ks1 = k1, ks2 = k0 ^ k1 ^ 0x1BD11BDAu;
  unsigned X0 = x0 + ks0, X1 = x1 + ks1;
  const int R0[4] = {13, 15, 26, 6};
  const int R1[4] = {17, 29, 16, 24};
#pragma unroll
  for (int i = 0; i < 4; i++) { X0 += X1; X1 = rotl32(X1, R0[i]); X1 ^= X0; }
  X0 += ks1; X1 += ks2 + 1u;
#pragma unroll
  for (int i = 0; i < 4; i++) { X0 += X1; X1 = rotl32(X1, R1[i]); X1 ^= X0; }
  X0 += ks2; X1 += ks0 + 2u;
#pragma unroll
  for (int i = 0; i < 4; i++) { X0 += X1; X1 = rotl32(X1, R0[i]); X1 ^= X0; }
  X0 += ks0; X1 += ks1 + 3u;
#pragma unroll
  for (int i = 0; i < 4; i++) { X0 += X1; X1 = rotl32(X1, R1[i]); X1 ^= X0; }
  X0 += ks1; X1 += ks2 + 4u;
#pragma unroll
  for (int i = 0; i < 4; i++) { X0 += X1; X1 = rotl32(X1, R0[i]); X1 ^= X0; }
  X0 += ks2; X1 += ks0 + 5u;
  *o0 = X0; *o1 = X1;
}

__device__ __forceinline__ float bits_to_unit(unsigned b) {
  // jax.random.uniform: (bits>>9)|0x3f800000 bitcast f32, minus 1.0  -> [0,1)
  return __uint_as_float((b >> 9) | 0x3F800000u) - 1.0f;
}

// h_sampled = (u < prob), u from threefry with key (0,42), counters (i, i+1024)
__global__ __launch_bounds__(256) void sample_kernel(const float* __restrict__ prob,
                                                     float* __restrict__ h) {
  int i = blockIdx.x * 256 + threadIdx.x;   // 0..1023
  unsigned o0, o1;
  threefry2x32(0u, 42u, (unsigned)i, (unsigned)(i + HWPIX), &o0, &o1);
  h[i]         = (bits_to_unit(o0) < prob[i])         ? 1.0f : 0.0f;
  h[i + HWPIX] = (bits_to_unit(o1) < prob[i + HWPIX]) ? 1.0f : 0.0f;
}

// ------------------ Gaussian blur as Y = G * X * G^T (WMMA) ----------------
// op: 0 -> X = src0 ; 2 -> X = src0*src1 ; 3 -> X = (src0-src1)^2
__global__ __launch_bounds__(128) void blur_kernel(const float* __restrict__ src0,
                                                   const float* __restrict__ src1,
                                                   float* __restrict__ dst,
                                                   int op, Gauss gk) {
  __shared__ float Xs[IMG][IMG + 1];
  __shared__ float T1[IMG][IMG + 1];
  __shared__ float Gm[IMG][IMG + 1];
  const int tid  = threadIdx.x;
  const int base = blockIdx.x * HWPIX;

  for (int i = tid; i < HWPIX; i += 128) {
    float a = src0[base + i];
    float v;
    if (op == 0)      v = a;
    else if (op == 2) v = a * src1[base + i];
    else { float d = a - src1[base + i]; v = d * d; }
    int r = i >> 5, cidx = i & 31;
    Xs[r][cidx] = v;
    int d = cidx - r;   // banded SAME-padded 1D Gaussian operator
    Gm[r][cidx] = (d >= -KRAD && d <= KRAD) ? gk.g[d + KRAD] : 0.0f;
  }
  __syncthreads();

  const int wave = tid >> 5, lane = tid & 31;
  const int lo = lane & 15, hi = lane >> 4;
  const int I = wave >> 1, J = wave & 1;         // 16x16 output tile coords
  const int m = I * 16 + lo, n = J * 16 + lo;

  // Stage 1: T1 = G * X   (A: lane->M, vgpr/half->K ; B mirrored; ISA 7.12.2)
  v8f acc = {0.f, 0.f, 0.f, 0.f, 0.f, 0.f, 0.f, 0.f};
#pragma unroll
  for (int kb = 0; kb < 8; kb++) {
    int k0 = kb * 4 + hi * 2;
    v2f a = { Gm[m][k0], Gm[m][k0 + 1] };
    v2f b = { Xs[k0][n], Xs[k0 + 1][n] };
    acc = __builtin_amdgcn_wmma_f32_16x16x4_f32(false, a, false, b,
                                                (short)0, acc, false, false);
  }
#pragma unroll
  for (int v = 0; v < 8; v++) T1[I * 16 + hi * 8 + v][n] = acc[v];
  __syncthreads();

  // Stage 2: Y = T1 * G^T  (B[k][n] = G[n][k])
  v8f acc2 = {0.f, 0.f, 0.f, 0.f, 0.f, 0.f, 0.f, 0.f};
#pragma unroll
  for (int kb = 0; kb < 8; kb++) {
    int k0 = kb * 4 + hi * 2;
    v2f a = { T1[m][k0], T1[m][k0 + 1] };
    v2f b = { Gm[n][k0], Gm[n][k0 + 1] };
    acc2 = __builtin_amdgcn_wmma_f32_16x16x4_f32(false, a, false, b,
                                                 (short)0, acc2, false, false);
  }
#pragma unroll
  for (int v = 0; v < 8; v++)
    dst[base + (I * 16 + hi * 8 + v) * IMG + n] = acc2[v];
}

// ---------------------- per-pixel reward contribution ----------------------
__device__ __forceinline__ float rp_eval(float muh, float muc, float sigh,
                                         float sigc, float sighc, float cc) {
  float l    = (2.f * muh * muc + C1F) / (muh * muh + muc * muc + C1F);
  float sq   = sqrtf(fmaxf(sigh * sigc, 0.f) + EPSF);
  float cm   = (2.f * sq + C2F) / (sigh + sigc + C2F);
  float sm   = (2.f * sighc + C2F) / (sq + C2F + EPSF);
  float ssim = l * cm * sm;
  float d    = muh - muc;
  return -(d * d) + WSF * (cc * ssim + (1.f - cc));
}

// Base per-pixel reward map + per-batch sum. (h binary => blur(h^2)=mu_h.)
__global__ __launch_bounds__(256) void base_kernel(const float* __restrict__ mu_h,
                                                   const float* __restrict__ mu_c,
                                                   const float* __restrict__ Shc,
                                                   const float* __restrict__ Sc2,
                                                   const float* __restrict__ c_var,
                                                   float* __restrict__ rp_base,
                                                   float* __restrict__ Rsum) {
  __shared__ float red[256];
  int b = blockIdx.x, tid = threadIdx.x;
  float s = 0.f;
#pragma unroll
  for (int i = 0; i < 4; i++) {
    int p = b * HWPIX + tid + i * 256;
    float muh = mu_h[p], muc = mu_c[p];
    float sigh  = muh - muh * muh;
    float sigc  = Sc2[p] - muc * muc;
    float sighc = Shc[p] - muh * muc;
    float cc = fminf(fmaxf(2.f * sqrtf(c_var[p] + EPSF), 0.f), 1.f);
    float r = rp_eval(muh, muc, sigh, sigc, sighc, cc);
    rp_base[p] = r;
    s += r;
  }
  red[tid] = s;
  __syncthreads();
  for (int off = 128; off > 0; off >>= 1) {
    if (tid < off) red[tid] += red[tid + off];
    __syncthreads();
  }
  if (tid == 0) Rsum[b] = red[0];
}

// One block per (b, pixel a): evaluate the flipped candidate's reward delta
// over the 11x11 affected window (rank-1 update of the blurred fields).
__global__ __launch_bounds__(128) void cand_kernel(const float* __restrict__ h,
                                                   const float* __restrict__ prob,
                                                   const float* __restrict__ c,
                                                   const float* __restrict__ mu_h,
                                                   const float* __restrict__ mu_c,
                                                   const float* __restrict__ Shc,
                                                   const float* __restrict__ Sc2,
                                                   const float* __restrict__ c_var,
                                                   const float* __restrict__ rp_base,
                                                   float* __restrict__ partial,
                                                   Gauss gk) {
  __shared__ float red[128];
  int b  = blockIdx.x >> 10;
  int a  = blockIdx.x & (HWPIX - 1);
  int ay = a >> 5, ax = a & 31;
  int tid = threadIdx.x;
  float ha = h[b * HWPIX + a];
  float pa = prob[b * HWPIX + a];
  float ca = c[b * HWPIX + a];
  float delta = 1.f - 2.f * ha;                 // flip direction
  float w = (ha > 0.5f) ? (1.f - pa) : pa;      // pi of the flipped action
  float val = 0.f;
  if (tid < KSIZE * KSIZE) {
    int dy = tid / KSIZE - KRAD;
    int dx = tid % KSIZE - KRAD;
    int py = ay + dy, px = ax + dx;
    if (py >= 0 && py < IMG && px >= 0 && px < IMG) {
      int p = b * HWPIX + py * IMG + px;
      float kk = delta * gk.g[dy + KRAD] * gk.g[dx + KRAD];
      float muh   = mu_h[p] + kk;               // blur(cand)
      float muc   = mu_c[p];
      float sigh  = muh - muh * muh;            // blur(cand^2)=blur(cand): binary
      float sigc  = Sc2[p] - muc * muc;
      float sighc = (Shc[p] + kk * ca) - muh * muc;
      float cc = fminf(fmaxf(2.f * sqrtf(c_var[p] + EPSF), 0.f), 1.f);
      val = rp_eval(muh, muc, sigh, sigc, sighc, cc) - rp_base[p];
    }
  }
  red[tid] = val;
  __syncthreads();
  for (int off = 64; off > 0; off >>= 1) {
    if (tid < off) red[tid] += red[tid + off];
    __syncthreads();
  }
  if (tid == 0) partial[blockIdx.x] = w * red[0];
}

// loss = -(Rsum0 + Rsum1 + P/HW) / (B*HW)   (unmatched actions give R_base;
// both action weights at a pixel sum to 1)
__global__ __launch_bounds__(256) void final_kernel(const float* __restrict__ partial,
                                                    const float* __restrict__ Rsum,
                                                    float* __restrict__ out) {
  __shared__ float red[256];
  int tid = threadIdx.x;
  float s = 0.f;
#pragma unroll
  for (int i = 0; i < 8; i++) s += partial[tid + i * 256];
  red[tid] = s;
  __syncthreads();
  for (int off = 128; off > 0; off >>= 1) {
    if (tid < off) red[tid] += red[tid + off];
    __syncthreads();
  }
  if (tid == 0) {
    float total = Rsum[0] + Rsum[1] + red[0] * (1.0f / (float)HWPIX);
    out[0] = -total / (float)(BATCH * HWPIX);
  }
}

// ---------------------------------------------------------------------------
extern "C" void kernel_launch(void* const* d_in, const int* in_sizes, int n_in,
                              void* d_out, int out_size, void* d_ws, size_t ws_size,
                              hipStream_t stream) {
  (void)in_sizes; (void)n_in; (void)out_size; (void)ws_size;
  const float* prob = (const float*)d_in[0];
  const float* c    = (const float*)d_in[1];
  // d_in[2] = z is unused by the reference loss.

  float* ws    = (float*)d_ws;
  float* h     = ws;            // 2048
  float* mu_h  = ws + 2048;     // 2048
  float* mu_c  = ws + 4096;     // 2048
  float* Sc2   = ws + 6144;     // 2048  blur(c*c)
  float* Shc   = ws + 8192;     // 2048  blur(h*c)
  float* c_var = ws + 10240;    // 2048  blur((c-mu_c)^2)
  float* rp    = ws + 12288;    // 2048  base per-pixel reward
  float* Rsum  = ws + 14336;    // 2
  float* part  = ws + 14344;    // 2048

  Gauss gk;
  {
    double t[KSIZE], s = 0.0;
    for (int i = 0; i < KSIZE; i++) {
      double r = (double)(i - KRAD);
      t[i] = exp(-0.5 * r * r / 4.0);   // sigma = 2.0
      s += t[i];
    }
    for (int i = 0; i < KSIZE; i++) gk.g[i] = (float)(t[i] / s);
  }

  sample_kernel<<<4, 256, 0, stream>>>(prob, h);

  blur_kernel<<<BATCH, 128, 0, stream>>>(h,    nullptr, mu_h,  0, gk);
  blur_kernel<<<BATCH, 128, 0, stream>>>(c,    nullptr, mu_c,  0, gk);
  blur_kernel<<<BATCH, 128, 0, stream>>>(c,    c,       Sc2,   2, gk);
  blur_kernel<<<BATCH, 128, 0, stream>>>(h,    c,       Shc,   2, gk);
  blur_kernel<<<BATCH, 128, 0, stream>>>(c,    mu_c,    c_var, 3, gk);  // needs mu_c (stream-ordered)

  base_kernel<<<BATCH, 256, 0, stream>>>(mu_h, mu_c, Shc, Sc2, c_var, rp, Rsum);

  cand_kernel<<<BATCH * HWPIX, 128, 0, stream>>>(h, prob, c, mu_h, mu_c, Shc,
                                                 Sc2, c_var, rp, part, gk);

  final_kernel<<<1, 256, 0, stream>>>(part, Rsum, (float*)d_out);
}